// DiffKGBase_12378095747628
// MI455X (gfx1250) — compile-verified
//
#include <hip/hip_runtime.h>
#include <hip/hip_bf16.h>
#include <stdint.h>

typedef __attribute__((ext_vector_type(2))) float v2f;
typedef __attribute__((ext_vector_type(8))) float v8f;

#define HOPS 3
#define B_   8
#define S_   256
#define H_   768
#define P_   256
#define NE   20000
#define NT   100000
#define NR   200
#define TTOK 4
#define KTOT (TTOK*H_)   /* 3072 */

/* GEMM tiling */
#define KC     32            /* K per staged chunk            */
#define NCHUNK (KTOT / KC)   /* 96                            */
#define BROW   36            /* padded row stride in floats:
                                144 B -> 16B-aligned rows and
                                bank-conflict-free frag reads  */

/* workspace layout (floats) */
#define OFF_WVP    0
#define OFF_D      (OFF_WVP + H_)
#define OFF_RS     (OFF_D + B_*H_)
#define OFF_CS     (OFF_RS + B_*HOPS*NR)
#define OFF_LD     (OFF_CS + B_*HOPS*2)
#define OFF_AREP   (OFF_LD + B_*H_)
#define OFF_SCORES (OFF_AREP + 16*KTOT)
#define OFF_WALK   (OFF_SCORES + B_*NE)
#define OFF_ECUR   (OFF_WALK + B_*HOPS*NE)
#define OFF_WALKED (OFF_ECUR + B_*NE)

/* -------- k0: wvp[h] = sum_p W_v[h,p] * W_p[p] -------- */
__global__ void k0_wvp(const float* __restrict__ Wv, const float* __restrict__ Wp,
                       float* __restrict__ ws) {
  int h = blockIdx.x * blockDim.x + threadIdx.x;
  if (h < H_) {
    float acc = 0.f;
    for (int p = 0; p < P_; ++p) acc += Wv[h * P_ + p] * Wp[p];
    ws[OFF_WVP + h] = acc;
  }
}

/* -------- k1: pointer softmax over S, then D[b,h] -------- */
__global__ void k1_pointer(const float* __restrict__ lhs, const float* __restrict__ mask,
                           float* __restrict__ ws) {
  __shared__ float red[S_];
  __shared__ float pbuf[S_];
  int b = blockIdx.x, s = threadIdx.x;
  const float* wvp = ws + OFF_WVP;
  const float* row = lhs + (size_t)(b * S_ + s) * H_;
  float pl = 0.f;
  for (int h = 0; h < H_; ++h) pl += row[h] * wvp[h];

  red[s] = pl; __syncthreads();
  for (int off = 128; off > 0; off >>= 1) {
    if (s < off) red[s] = fmaxf(red[s], red[s + off]);
    __syncthreads();
  }
  float mx = red[0]; __syncthreads();
  float e = expf(pl - mx);
  red[s] = e; __syncthreads();
  for (int off = 128; off > 0; off >>= 1) {
    if (s < off) red[s] += red[s + off];
    __syncthreads();
  }
  float sum = red[0];
  pbuf[s] = (e / sum) * mask[b * S_ + s];
  __syncthreads();

  for (int h = s; h < H_; h += S_) {
    float acc = 0.f;
    for (int s2 = 0; s2 < S_; ++s2)
      acc += pbuf[s2] * lhs[(size_t)(b * S_ + s2) * H_ + h];
    ws[OFF_D + b * H_ + h] = acc;
  }
}

/* -------- k2: rels_seq / checks_seq / LD -------- */
__global__ void k2_proj(const float* __restrict__ Wr, const float* __restrict__ Wc,
                        const float* __restrict__ Lw, float* __restrict__ ws) {
  __shared__ float Dl[H_];
  __shared__ float rl[HOPS * NR];
  __shared__ float cl[HOPS * 2];
  int b = blockIdx.x, t = threadIdx.x;
  for (int h = t; h < H_; h += 256) Dl[h] = ws[OFF_D + b * H_ + h];
  __syncthreads();

  for (int o = t; o < HOPS * NR; o += 256) {
    float acc = 0.f;
    for (int h = 0; h < H_; ++h) acc += Dl[h] * Wr[(size_t)h * (HOPS * NR) + o];
    rl[o] = acc;
  }
  if (t < HOPS * 2) {
    float acc = 0.f;
    for (int h = 0; h < H_; ++h) acc += Dl[h] * Wc[h * (HOPS * 2) + t];
    cl[t] = acc;
  }
  __syncthreads();

  if (t < HOPS) {
    float mx = -1e30f;
    for (int i = 0; i < NR; ++i) mx = fmaxf(mx, rl[t * NR + i]);
    float sm = 0.f;
    for (int i = 0; i < NR; ++i) sm += expf(rl[t * NR + i] - mx);
    for (int i = 0; i < NR; ++i)
      ws[OFF_RS + b * HOPS * NR + t * NR + i] = expf(rl[t * NR + i] - mx) / sm;

    float c0 = cl[t * 2], c1 = cl[t * 2 + 1];
    float m2 = fmaxf(c0, c1);
    float e0 = expf(c0 - m2), e1 = expf(c1 - m2);
    ws[OFF_CS + b * HOPS * 2 + t * 2 + 0] = e0 / (e0 + e1);
    ws[OFF_CS + b * HOPS * 2 + t * 2 + 1] = e1 / (e0 + e1);
  }

  for (int o = t; o < H_; o += 256) {
    float acc = 0.f;
    for (int h = 0; h < H_; ++h) acc += Dl[h] * Lw[(size_t)h * H_ + o];
    ws[OFF_LD + b * H_ + o] = acc;
  }
}

/* -------- k3: build replicated/padded A (16 x 3072) -------- */
__global__ void k3_arep(float* __restrict__ ws) {
  int idx = blockIdx.x * 256 + threadIdx.x;
  if (idx < 16 * KTOT) {
    int m = idx / KTOT, k = idx % KTOT;
    ws[OFF_AREP + idx] = (m < B_) ? ws[OFF_LD + m * H_ + (k % H_)] : 0.f;
  }
}

/* -------- k4: 3-hop relation walk (one block per batch) -------- */
__global__ void k4_walk(const float* __restrict__ init_ent, const int* __restrict__ heads,
                        const int* __restrict__ rels, const int* __restrict__ tails,
                        float* __restrict__ ws) {
  __shared__ float rs[HOPS * NR];
  __shared__ float red[512];
  int b = blockIdx.x, t = threadIdx.x;
  float* ecur = ws + OFF_ECUR + (size_t)b * NE;
  float* wk   = ws + OFF_WALKED + (size_t)b * NE;
  for (int i = t; i < NE; i += 512) ecur[i] = init_ent[(size_t)b * NE + i];
  for (int i = t; i < HOPS * NR; i += 512) rs[i] = ws[OFF_RS + b * HOPS * NR + i];
  __syncthreads();

  const int* hb = heads + (size_t)b * NT;
  const int* rb = rels  + (size_t)b * NT;
  const int* tb = tails + (size_t)b * NT;

  for (int hop = 0; hop < HOPS; ++hop) {
    for (int i = t; i < NE; i += 512) wk[i] = 0.f;
    __syncthreads();
    for (int i = t; i < NT; i += 512) {
      float trip = rs[hop * NR + rb[i]] * ecur[hb[i]];
      atomicAdd(&wk[tb[i]], trip);
    }
    __syncthreads();
    float part = 0.f;
    for (int i = t; i < NE; i += 512) part += wk[i];
    red[t] = part; __syncthreads();
    for (int off = 256; off > 0; off >>= 1) {
      if (t < off) red[t] += red[t + off];
      __syncthreads();
    }
    float inv = 1.0f / (red[0] + 1e-6f);
    __syncthreads();
    for (int i = t; i < NE; i += 512) {
      float w = wk[i] * inv;
      ws[OFF_WALK + ((size_t)b * HOPS + hop) * NE + i] = w;
      ecur[i] = w;
    }
    __syncthreads();
  }
}

/* -------- k5: scores = Arep(16x3072) @ E^T via V_WMMA_F32_16X16X4_F32 --------
   8 waves/block, 16 entities per wave (128/block). Double-buffered LDS tiles:
   chunk k+1 streamed in with global_load_async_to_lds_b128 (ASYNCcnt) while
   WMMAs consume chunk k. Row stride padded to 36 floats so the 16 lanes of a
   fragment read land on 16 distinct LDS banks. One barrier per chunk. */
__global__ void __launch_bounds__(256) k5_scores(const float* __restrict__ ents,
                                                 float* __restrict__ ws) {
  __shared__ __align__(16) float Ach[2][16 * BROW];    /* 2 x 2.25 KB */
  __shared__ __align__(16) float Bch[2][128 * BROW];   /* 2 x 18 KB   */
  int tid  = threadIdx.x;
  int wave = tid >> 5, lane = tid & 31;
  int e0 = blockIdx.x * 128;
  const float* Arep = ws + OFF_AREP;

  v8f c = {0.f, 0.f, 0.f, 0.f, 0.f, 0.f, 0.f, 0.f};

  unsigned b0base = (unsigned)(uintptr_t)(&Bch[0][0]);  /* LDS byte offsets */
  unsigned b1base = (unsigned)(uintptr_t)(&Bch[1][0]);
  unsigned long long ebase = (unsigned long long)(uintptr_t)ents;

  int m    = lane & 15;
  int ksel = (lane < 16) ? 0 : 2;

  /* per-thread staging coordinates (constant across chunks) */
  int aj = tid * 2;                  /* A: 512 floats, 2/thread  */
  int am = aj >> 5, ak = aj & 31;

  /* ---- stage helpers (macros keep asm operands simple) ---- */
#define STAGE_A(buf, k0)                                                     \
  {                                                                          \
    v2f a2 = *(const v2f*)(Arep + (size_t)am * KTOT + (k0) + ak);            \
    *(v2f*)&Ach[buf][am * BROW + ak] = a2;                                   \
  }

#define STAGE_B_ASYNC(bufbase, k0)                                           \
  for (int it = 0; it < 4; ++it) {                                           \
    int q   = it * 256 + tid;                                                \
    int row = q >> 3, seg = q & 7;                                           \
    int er  = e0 + row; if (er >= NE) er = NE - 1;                           \
    unsigned goff = (unsigned)(((unsigned long long)er * KTOT +              \
                                (unsigned)(k0)) * 4u + (unsigned)seg * 16u); \
    unsigned ldsa = (bufbase) + (unsigned)(row * (BROW * 4) + seg * 16);     \
    asm volatile("global_load_async_to_lds_b128 %0, %1, %2"                  \
                 :: "v"(ldsa), "v"(goff), "s"(ebase) : "memory");            \
  }

  /* prologue: fill buffer 0 with chunk 0 */
  STAGE_A(0, 0)
  STAGE_B_ASYNC(b0base, 0)
  asm volatile("s_wait_asynccnt 0" ::: "memory");
  __syncthreads();

  for (int kc = 0; kc < NCHUNK; ++kc) {
    int cur = kc & 1;
    /* kick off next chunk into the other buffer (overlaps with compute) */
    if (kc + 1 < NCHUNK) {
      int k1 = (kc + 1) * KC;
      if (cur == 0) { STAGE_A(1, k1) STAGE_B_ASYNC(b1base, k1) }
      else          { STAGE_A(0, k1) STAGE_B_ASYNC(b0base, k1) }
    }

    /* 8 WMMA f32 16x16x4 steps over the current chunk */
    const float* Ab = &Ach[cur][0];
    const float* Bb = &Bch[cur][(wave * 16) * BROW];
#pragma unroll
    for (int kk = 0; kk < KC; kk += 4) {
      v2f a  = *(const v2f*)&Ab[m * BROW + kk + ksel];
      v2f bf = *(const v2f*)&Bb[m * BROW + kk + ksel];
      c = __builtin_amdgcn_wmma_f32_16x16x4_f32(false, a, false, bf,
                                                (short)0, c, false, false);
    }

    /* next buffer fully written + everyone done reading current */
    asm volatile("s_wait_asynccnt 0" ::: "memory");
    __syncthreads();
  }

  /* C layout: VGPR v -> M=v (lanes 0-15) / M=v+8 (lanes 16-31); N=lane&15 */
  if (lane < 16) {
    int e = e0 + wave * 16 + lane;
    if (e < NE) {
#pragma unroll
      for (int v = 0; v < 8; ++v) ws[OFF_SCORES + (size_t)v * NE + e] = c[v];
    }
  }
#undef STAGE_A
#undef STAGE_B_ASYNC
}

/* -------- k6: checks softmax over N_E + combine with checks_seq -------- */
__global__ void k6_final(float* __restrict__ ws, float* __restrict__ out) {
  __shared__ float red[256];
  int blk = blockIdx.x, t = threadIdx.x;
  int b = blk / HOPS, hop = blk % HOPS;
  const float* wlk = ws + OFF_WALK + ((size_t)b * HOPS + hop) * NE;
  const float* sc  = ws + OFF_SCORES + (size_t)b * NE;
  float cs0 = ws[OFF_CS + b * HOPS * 2 + hop * 2 + 0];
  float cs1 = ws[OFF_CS + b * HOPS * 2 + hop * 2 + 1];

  float mx = -1e30f;
  for (int i = t; i < NE; i += 256) mx = fmaxf(mx, wlk[i] * sc[i]);
  red[t] = mx; __syncthreads();
  for (int off = 128; off > 0; off >>= 1) {
    if (t < off) red[t] = fmaxf(red[t], red[t + off]);
    __syncthreads();
  }
  mx = red[0]; __syncthreads();

  float sm = 0.f;
  for (int i = t; i < NE; i += 256) sm += expf(wlk[i] * sc[i] - mx);
  red[t] = sm; __syncthreads();
  for (int off = 128; off > 0; off >>= 1) {
    if (t < off) red[t] += red[t + off];
    __syncthreads();
  }
  sm = red[0];

  float* ob = out + ((size_t)b * HOPS + hop) * NE;
  for (int i = t; i < NE; i += 256) {
    float x = wlk[i] * sc[i];
    ob[i] = cs0 * wlk[i] + cs1 * (expf(x - mx) / sm);
  }
}

extern "C" void kernel_launch(void* const* d_in, const int* in_sizes, int n_in,
                              void* d_out, int out_size, void* d_ws, size_t ws_size,
                              hipStream_t stream) {
  const float* lhs      = (const float*)d_in[0];
  const float* mask     = (const float*)d_in[1];
  const float* init_ent = (const float*)d_in[2];
  const float* ents     = (const float*)d_in[3];
  /* d_in[4] = W_q: unused — its contribution is constant over S and cancels
     inside the softmax over S. */
  const float* Wv       = (const float*)d_in[5];
  const float* Wp       = (const float*)d_in[6];
  const float* Wr       = (const float*)d_in[7];
  const float* Wc       = (const float*)d_in[8];
  const float* Lw       = (const float*)d_in[9];
  const int*   heads    = (const int*)d_in[10];
  const int*   rels     = (const int*)d_in[11];
  const int*   tails    = (const int*)d_in[12];
  float* ws  = (float*)d_ws;
  float* out = (float*)d_out;
  (void)in_sizes; (void)n_in; (void)out_size; (void)ws_size;

  hipLaunchKernelGGL(k0_wvp,    dim3(3),        dim3(256), 0, stream, Wv, Wp, ws);
  hipLaunchKernelGGL(k1_pointer,dim3(B_),       dim3(S_),  0, stream, lhs, mask, ws);
  hipLaunchKernelGGL(k2_proj,   dim3(B_),       dim3(256), 0, stream, Wr, Wc, Lw, ws);
  hipLaunchKernelGGL(k3_arep,   dim3(192),      dim3(256), 0, stream, ws);
  hipLaunchKernelGGL(k4_walk,   dim3(B_),       dim3(512), 0, stream, init_ent, heads, rels, tails, ws);
  hipLaunchKernelGGL(k5_scores, dim3((NE + 127) / 128), dim3(256), 0, stream, ents, ws);
  hipLaunchKernelGGL(k6_final,  dim3(B_ * HOPS), dim3(256), 0, stream, ws, out);
}